// Sample_generator_48017734369828
// MI455X (gfx1250) — compile-verified
//
#include <hip/hip_runtime.h>
#include <stdint.h>

// Problem dims (from reference setup_inputs)
#define B_   32
#define T_   4096
#define F_   1024
#define HARDK 10
#define NORK  5
#define ABNK  10

// Flat float offsets of the 8 outputs in d_out (return order)
#define OFF_FEAT_NOR   0          // 32*5*1024   = 163840
#define OFF_SCORE_NOR  163840     // 32*5        = 160
#define OFF_IDX_NOR    164000     // 32*5        = 160
#define OFF_FEAT_ABN   164160     // 32*10*1024  = 327680
#define OFF_SCORE_ABN  491840     // 32*10       = 320
#define OFF_FEAT_HARD  492160     // 32*10*1024  = 327680
#define OFF_SCORE_HARD 819840     // 32*10       = 320
#define OFF_IDX_HARD   820160     // 32*10       = 320

// Insert `key` into an ascending sorted list of K uint64 keys (min-K kept).
// Fully unrolled single bubble pass -> constant indices -> stays in VGPRs.
template <int K>
__device__ __forceinline__ void insert_key(uint64_t (&lst)[K], uint64_t key) {
  if (key < lst[K - 1]) lst[K - 1] = key;
#pragma unroll
  for (int i = K - 1; i > 0; --i) {
    uint64_t a = lst[i - 1];
    uint64_t c = lst[i];
    if (c < a) { lst[i - 1] = c; lst[i] = a; }
  }
}

// ---------------------------------------------------------------------------
// Kernel 1: one wave (32 lanes) per batch row. Single pass over 4096 scores,
// maintaining three register top-k lists per lane; LDS merge by lane 0.
// key64 = (monotone_key_bits << 32) | index  (minimize; tie -> lower index)
// ---------------------------------------------------------------------------
__global__ __launch_bounds__(32) void topk_kernel(const float* __restrict__ score,
                                                  float* __restrict__ out,
                                                  int* __restrict__ ws_idx) {
  const int b = blockIdx.x;
  const unsigned lane = threadIdx.x;
  const float* srow = score + (size_t)b * T_;

  uint64_t lh[HARDK], ln[NORK], la[ABNK];
#pragma unroll
  for (int k = 0; k < HARDK; ++k) lh[k] = ~0ull;
#pragma unroll
  for (int k = 0; k < NORK; ++k) ln[k] = ~0ull;
#pragma unroll
  for (int k = 0; k < ABNK; ++k) la[k] = ~0ull;

  for (int t = lane; t < T_; t += 32) {
    float s = srow[t];
    unsigned sb = __float_as_uint(s);                  // s in [0,1): monotone bits
    unsigned hb = __float_as_uint(fabsf(s - 0.5f));    // >= 0: monotone bits
    insert_key(lh, ((uint64_t)hb << 32) | (unsigned)t);                   // min |s-0.5|
    insert_key(ln, ((uint64_t)sb << 32) | (unsigned)t);                   // min s
    insert_key(la, ((uint64_t)(0xFFFFFFFFu - sb) << 32) | (unsigned)t);   // max s
  }

  __shared__ uint64_t sh[32 * HARDK];
  __shared__ uint64_t sn[32 * NORK];
  __shared__ uint64_t sa[32 * ABNK];
#pragma unroll
  for (int k = 0; k < HARDK; ++k) sh[lane * HARDK + k] = lh[k];
#pragma unroll
  for (int k = 0; k < NORK; ++k) sn[lane * NORK + k] = ln[k];
#pragma unroll
  for (int k = 0; k < ABNK; ++k) sa[lane * ABNK + k] = la[k];
  __syncthreads();

  if (lane == 0) {
    uint64_t fh[HARDK], fn[NORK], fa[ABNK];
#pragma unroll
    for (int k = 0; k < HARDK; ++k) fh[k] = ~0ull;
#pragma unroll
    for (int k = 0; k < NORK; ++k) fn[k] = ~0ull;
#pragma unroll
    for (int k = 0; k < ABNK; ++k) fa[k] = ~0ull;

    for (int i = 0; i < 32 * HARDK; ++i) insert_key(fh, sh[i]);
    for (int i = 0; i < 32 * NORK; ++i)  insert_key(fn, sn[i]);
    for (int i = 0; i < 32 * ABNK; ++i)  insert_key(fa, sa[i]);

    // conf_nor: slots [0..4]
    for (int k = 0; k < NORK; ++k) {
      int idx = (int)(fn[k] & 0xFFFFFFFFu);
      out[OFF_SCORE_NOR + b * NORK + k] = srow[idx];
      out[OFF_IDX_NOR + b * NORK + k] = (float)idx;
      ws_idx[b * 25 + k] = idx;
    }
    // conf_abn: slots [5..14]
    for (int k = 0; k < ABNK; ++k) {
      int idx = (int)(fa[k] & 0xFFFFFFFFu);
      out[OFF_SCORE_ABN + b * ABNK + k] = srow[idx];
      ws_idx[b * 25 + 5 + k] = idx;
    }
    // hard: slots [15..24]
    for (int k = 0; k < HARDK; ++k) {
      int idx = (int)(fh[k] & 0xFFFFFFFFu);
      out[OFF_SCORE_HARD + b * HARDK + k] = srow[idx];
      out[OFF_IDX_HARD + b * HARDK + k] = (float)idx;
      ws_idx[b * 25 + 15 + k] = idx;
    }
  }
}

// ---------------------------------------------------------------------------
// Kernel 2: index-driven row gather via the CDNA5 async LDS data mover.
// One wave per output row (4 KB). 8x global_load_async_to_lds_b128 per lane
// into a 4 KB LDS staging buffer, s_wait_asynccnt 0, then 8x
// global_store_async_from_lds_b128 to the destination row.
// ---------------------------------------------------------------------------
__global__ __launch_bounds__(32) void gather_kernel(const float* __restrict__ feat,
                                                    const int* __restrict__ ws_idx,
                                                    float* __restrict__ out) {
  __shared__ __align__(16) unsigned char smem[4096];

  const int r = blockIdx.x;          // 0..799
  const int b = r / 25;
  const int slot = r % 25;
  const int idx = ws_idx[r];

  const float* src = feat + ((size_t)b * T_ + (size_t)idx) * F_;
  float* dst;
  if (slot < 5)
    dst = out + OFF_FEAT_NOR + ((size_t)b * NORK + slot) * F_;
  else if (slot < 15)
    dst = out + OFF_FEAT_ABN + ((size_t)b * ABNK + (slot - 5)) * F_;
  else
    dst = out + OFF_FEAT_HARD + ((size_t)b * ABNK + (slot - 15)) * F_;

  const unsigned lane = threadIdx.x;
  const unsigned ldsbase = (unsigned)(uintptr_t)(&smem[0]);   // low 32 bits = LDS offset

  // 8 outstanding async 16B loads per lane: 32 lanes * 8 * 16B = 4 KB row
#pragma unroll
  for (int j = 0; j < 8; ++j) {
    unsigned off = lane * 16u + (unsigned)j * 512u;
    uint64_t ga = (uint64_t)(uintptr_t)src + off;
    unsigned la = ldsbase + off;
    asm volatile("global_load_async_to_lds_b128 %0, %1, off"
                 :
                 : "v"(la), "v"(ga)
                 : "memory");
  }
  asm volatile("s_wait_asynccnt 0x0" ::: "memory");

#pragma unroll
  for (int j = 0; j < 8; ++j) {
    unsigned off = lane * 16u + (unsigned)j * 512u;
    uint64_t ga = (uint64_t)(uintptr_t)dst + off;
    unsigned la = ldsbase + off;
    asm volatile("global_store_async_from_lds_b128 %0, %1, off"
                 :
                 : "v"(ga), "v"(la)
                 : "memory");
  }
  asm volatile("s_wait_asynccnt 0x0" ::: "memory");
}

extern "C" void kernel_launch(void* const* d_in, const int* in_sizes, int n_in,
                              void* d_out, int out_size, void* d_ws, size_t ws_size,
                              hipStream_t stream) {
  const float* feat  = (const float*)d_in[0];   // [32, 4096, 1024] f32
  const float* score = (const float*)d_in[1];   // [32, 4096] f32
  float* out = (float*)d_out;                   // 820480 floats
  int* ws_idx = (int*)d_ws;                     // 800 ints of scratch

  topk_kernel<<<B_, 32, 0, stream>>>(score, out, ws_idx);
  gather_kernel<<<B_ * 25, 32, 0, stream>>>(feat, ws_idx, out);
}